// OptNet_41970420417254
// MI455X (gfx1250) — compile-verified
//
#include <hip/hip_runtime.h>
#include <math.h>

// ---- problem sizes -------------------------------------------------------
#define NF    128      // features (output cols)
#define NH    255      // hidden / QP dimension n
#define NI    254      // inequality count m
#define NB    128      // batch
#define NP    256      // padded dimension (both n and m padded to 256)
#define KST   257      // LDS row stride for K (bank-conflict-free columns)
#define EPSQ  1e-4f
#define SIG   0.1f
#define NITER 20

typedef __attribute__((ext_vector_type(2))) float v2f;
typedef __attribute__((ext_vector_type(8))) float v8f;

// ---- setup kernels -------------------------------------------------------

// z = 0, s = 1 (pad s = 1), lam = 1 (pad lam = 0)
__global__ void k_init(float* __restrict__ zg, float* __restrict__ sg,
                       float* __restrict__ lamg) {
  int b = blockIdx.x, i = threadIdx.x;
  zg[b * NP + i]   = 0.f;
  sg[b * NP + i]   = 1.f;
  lamg[b * NP + i] = (i < NI) ? 1.f : 0.f;
}

// Qp[i][j] = sum_{k<=min(i,j)} L[i][k]*L[j][k] + eps*(i==j)   (i,j < NH)
// padding: Qp[255][255] = 1, rest of pad = 0  (keeps padded K SPD)
__global__ void k_qbuild(const float* __restrict__ L, float* __restrict__ Qp) {
  int i = blockIdx.x, j = threadIdx.x;
  float acc;
  if (i < NH && j < NH) {
    acc = (i == j) ? EPSQ : 0.f;
    int kmax = (i < j) ? i : j;
    const float* ri = L + i * NH;
    const float* rj = L + j * NH;
    for (int k = 0; k <= kmax; ++k) acc = fmaf(ri[k], rj[k], acc);
  } else {
    acc = (i == j) ? 1.f : 0.f;
  }
  Qp[i * NP + j] = acc;
}

// padded G (row-major) and its transpose
__global__ void k_gbuild(const float* __restrict__ G, float* __restrict__ Gp,
                         float* __restrict__ GT) {
  int m = blockIdx.x, i = threadIdx.x;
  float v = (m < NI && i < NH) ? G[m * NH + i] : 0.f;
  Gp[m * NP + i] = v;
  GT[i * NP + m] = v;
}

// p[b][j] = sum_k x[b][k] * W[j][k] + bias[j]
__global__ void k_pbuild(const float* __restrict__ x, const float* __restrict__ W,
                         const float* __restrict__ bias, float* __restrict__ pg) {
  int b = blockIdx.x, j = threadIdx.x;
  float acc = 0.f;
  if (j < NH) {
    acc = bias[j];
    const float* xr = x + b * NF;
    const float* wr = W + j * NF;
    for (int k = 0; k < NF; ++k) acc = fmaf(xr[k], wr[k], acc);
  }
  pg[b * NP + j] = acc;
}

// h[m] = G[m,:] . z0 + s0[m]
__global__ void k_hbuild(const float* __restrict__ G, const float* __restrict__ z0,
                         const float* __restrict__ s0, float* __restrict__ hg) {
  int m = threadIdx.x;
  float acc = 0.f;
  if (m < NI) {
    acc = s0[m];
    const float* gr = G + m * NH;
    for (int i = 0; i < NH; ++i) acc = fmaf(gr[i], z0[i], acc);
  }
  hg[m] = acc;
}

// ---- fused interior-point iteration: one workgroup per batch row ---------
// LDS: K (256x257 f32 = 263KB) + 11 vectors of 256 f32 -> ~274KB (< 320KB/WGP)
__global__ __launch_bounds__(256) void k_iter(
    float* __restrict__ zg, float* __restrict__ sg, float* __restrict__ lamg,
    const float* __restrict__ pg, const float* __restrict__ Qg,
    const float* __restrict__ Gp, const float* __restrict__ GT,
    const float* __restrict__ hg) {
  extern __shared__ float lds[];
  float* Ksh   = lds;                 // KST*NP
  float* zsh   = lds + KST * NP;      // 256 each below
  float* ssh   = zsh + NP;
  float* lamsh = ssh + NP;
  float* dsh   = lamsh + NP;
  float* g2sh  = dsh + NP;
  float* r2sh  = g2sh + NP;
  float* r3sh  = r2sh + NP;
  float* rhs   = r3sh + NP;           // becomes y during fwd solve
  float* dzsh  = rhs + NP;
  float* invd  = dzsh + NP;
  float* red   = invd + NP;

  const int b   = blockIdx.x;
  const int tid = threadIdx.x;
  const bool mok = tid < NI;

  const int lane  = tid & 31;
  const int wave  = tid >> 5;
  const int nlane = lane & 15;         // M for A, N (col) for B/C
  const int koff  = (lane >> 4) << 1;  // 0 for lanes 0-15, 2 for lanes 16-31
  const int half8 = (lane >> 4) << 3;  // C/D row offset for lanes 16-31

  // --- load state ---------------------------------------------------------
  zsh[tid]   = zg[b * NP + tid];
  ssh[tid]   = sg[b * NP + tid];
  lamsh[tid] = lamg[b * NP + tid];
  __syncthreads();

  // --- r2[m] = z . G[m,:] + s - h ----------------------------------------
  float r2 = 0.f;
  if (mok) {
    float acc = 0.f;
    const float* gr = Gp + tid * NP;
    for (int i = 0; i < NP; ++i) acc = fmaf(zsh[i], gr[i], acc);
    r2 = acc + ssh[tid] - hg[tid];
  }
  r2sh[tid] = r2;

  // --- mu = mean(s*lam) over valid m (block sum reduction) ----------------
  red[tid] = mok ? ssh[tid] * lamsh[tid] : 0.f;
  __syncthreads();
  for (int off = NP / 2; off > 0; off >>= 1) {
    if (tid < off) red[tid] += red[tid + off];
    __syncthreads();
  }
  const float mu = red[0] * (1.f / (float)NI);

  // --- r3, d = lam/s, g2 = (lam*r2 - r3)/s (padding forced to 0) ----------
  float r3 = mok ? (ssh[tid] * lamsh[tid] - SIG * mu) : 0.f;
  r3sh[tid] = r3;
  dsh[tid]  = mok ? (lamsh[tid] / ssh[tid]) : 0.f;
  g2sh[tid] = mok ? ((lamsh[tid] * r2 - r3) / ssh[tid]) : 0.f;
  __syncthreads();

  // --- rhs[i] = -( z.Q[:,i] + p[i] + sum_m (lam_m+g2_m) G[m,i] ) ----------
  {
    float a1 = 0.f, a2 = 0.f;
    for (int j = 0; j < NP; ++j) a1 = fmaf(zsh[j], Qg[j * NP + tid], a1);
    for (int m = 0; m < NP; ++m)
      a2 = fmaf(lamsh[m] + g2sh[m], Gp[m * NP + tid], a2);
    rhs[tid] = -(a1 + pg[b * NP + tid] + a2);
  }
  __syncthreads();

  // --- K = Q + G^T diag(d) G via V_WMMA_F32_16X16X4_F32, built in LDS -----
  for (int t = wave; t < 256; t += 8) {
    const int i0  = (t >> 4) << 4;
    const int j0  = (t & 15) << 4;
    const int col = j0 + nlane;
    v8f c = {};
#pragma unroll 4
    for (int k0 = 0; k0 < NP; k0 += 4) {
      const int ka = k0 + koff;
      __builtin_prefetch(GT + (i0 + nlane) * NP + ka + 32, 0, 0);
      v2f a, bb;
      a.x  = GT[(i0 + nlane) * NP + ka];
      a.y  = GT[(i0 + nlane) * NP + ka + 1];
      bb.x = dsh[ka]     * Gp[ka * NP + col];
      bb.y = dsh[ka + 1] * Gp[(ka + 1) * NP + col];
#if __has_builtin(__builtin_amdgcn_wmma_f32_16x16x4_f32)
      c = __builtin_amdgcn_wmma_f32_16x16x4_f32(
          false, a, false, bb, (short)0, c, false, false);
#else
      for (int v = 0; v < 8; ++v)
        for (int kk = 0; kk < 4; ++kk)
          c[v] = fmaf(GT[(i0 + v + half8) * NP + k0 + kk],
                      dsh[k0 + kk] * Gp[(k0 + kk) * NP + col], c[v]);
#endif
    }
    // C/D layout: VGPR v holds M = v (+8 for lanes 16-31), N = lane&15
    for (int v = 0; v < 8; ++v) {
      const int r = i0 + half8 + v;
      Ksh[r * KST + col] = c[v] + Qg[r * NP + col];
    }
  }
  __syncthreads();

  // --- blocked in-LDS Cholesky (panel = 16 columns) -----------------------
  // Panel factorization: serial column sweep restricted to panel columns.
  // Trailing rank-16 update: WMMA on 16x16 tiles (lower triangle only).
  for (int p = 0; p < 16; ++p) {
    const int k0 = p << 4;
    // panel factorization (columns k0 .. k0+15, full height below diagonal)
    for (int kk = 0; kk < 16; ++kk) {
      const int k = k0 + kk;
      if (tid == 0) {
        float dkk = Ksh[k * KST + k];
        dkk = (dkk > 1e-20f) ? sqrtf(dkk) : 1e-10f;
        Ksh[k * KST + k] = dkk;
        invd[k] = 1.f / dkk;
      }
      __syncthreads();
      const int i = k + 1 + tid;
      if (i < NP) Ksh[i * KST + k] *= invd[k];
      __syncthreads();
      if (i < NP) {
        const float lik = Ksh[i * KST + k];
        const int jend = k0 + 16;
        for (int j = k + 1; j < jend; ++j)
          Ksh[i * KST + j] = fmaf(-lik, Ksh[j * KST + k], Ksh[i * KST + j]);
      }
      __syncthreads();
    }
    // trailing update: C(ti,tj) -= Lpanel(ti) * Lpanel(tj)^T  via WMMA
    int idx = 0;
    for (int ti = p + 1; ti < 16; ++ti) {
      for (int tj = p + 1; tj <= ti; ++tj, ++idx) {
        if ((idx & 7) != wave) continue;     // wave-uniform
        const int i0 = ti << 4, j0 = tj << 4;
        v8f c;
        for (int v = 0; v < 8; ++v)
          c[v] = Ksh[(i0 + half8 + v) * KST + j0 + nlane];
#pragma unroll
        for (int kq = 0; kq < 16; kq += 4) {
          const int ka = k0 + kq + koff;
          v2f a, bb;
          // A[m][k] = L[i0+m][ka]
          a.x  =  Ksh[(i0 + nlane) * KST + ka];
          a.y  =  Ksh[(i0 + nlane) * KST + ka + 1];
          // B[k][n] = L[j0+n][ka], negated to realize C -= A*B
          bb.x = -Ksh[(j0 + nlane) * KST + ka];
          bb.y = -Ksh[(j0 + nlane) * KST + ka + 1];
#if __has_builtin(__builtin_amdgcn_wmma_f32_16x16x4_f32)
          c = __builtin_amdgcn_wmma_f32_16x16x4_f32(
              false, a, false, bb, (short)0, c, false, false);
#else
          for (int v = 0; v < 8; ++v)
            for (int q = 0; q < 4; ++q)
              c[v] = fmaf(Ksh[(i0 + half8 + v) * KST + k0 + kq + q],
                          -Ksh[(j0 + nlane) * KST + k0 + kq + q], c[v]);
#endif
        }
        for (int v = 0; v < 8; ++v)
          Ksh[(i0 + half8 + v) * KST + j0 + nlane] = c[v];
      }
    }
    __syncthreads();
  }

  // --- forward solve L y = rhs (in place in rhs) --------------------------
  for (int k = 0; k < NP; ++k) {
    if (tid == 0) rhs[k] *= invd[k];
    __syncthreads();
    const float yk = rhs[k];
    const int i = k + 1 + tid;
    if (i < NP) rhs[i] = fmaf(-Ksh[i * KST + k], yk, rhs[i]);
    __syncthreads();
  }
  // --- backward solve L^T dz = y ------------------------------------------
  for (int k = NP - 1; k >= 0; --k) {
    if (tid == 0) dzsh[k] = rhs[k] * invd[k];
    __syncthreads();
    const float xk = dzsh[k];
    if (tid < k) rhs[tid] = fmaf(-Ksh[k * KST + tid], xk, rhs[tid]);
    __syncthreads();
  }

  // --- ds, dlam, per-row step ratios --------------------------------------
  float ds = 0.f, dlam = 0.f, ratio = 1e10f;
  if (mok) {
    float a3 = 0.f;
    const float* gr = Gp + tid * NP;
    for (int i = 0; i < NP; ++i) a3 = fmaf(dzsh[i], gr[i], a3);
    ds   = -r2sh[tid] - a3;
    dlam = (-r3sh[tid] - lamsh[tid] * ds) / ssh[tid];
    const float rs = (ds   < 0.f) ? (-ssh[tid]   / ds)   : 1e10f;
    const float rl = (dlam < 0.f) ? (-lamsh[tid] / dlam) : 1e10f;
    ratio = fminf(rs, rl);
  }
  __syncthreads();
  red[tid] = ratio;
  __syncthreads();
  for (int off = NP / 2; off > 0; off >>= 1) {
    if (tid < off) red[tid] = fminf(red[tid], red[tid + off]);
    __syncthreads();
  }
  const float alpha = fminf(1.f, 0.99f * red[0]);

  // --- state update --------------------------------------------------------
  zg[b * NP + tid] = zsh[tid] + alpha * dzsh[tid];
  if (mok) {
    sg[b * NP + tid]   = ssh[tid]   + alpha * ds;
    lamg[b * NP + tid] = lamsh[tid] + alpha * dlam;
  }
}

// out[b][i] = z[b][i], i < NF
__global__ void k_out(const float* __restrict__ zg, float* __restrict__ out) {
  int b = blockIdx.x, i = threadIdx.x;
  out[b * NF + i] = zg[b * NP + i];
}

// ---- host launcher -------------------------------------------------------
extern "C" void kernel_launch(void* const* d_in, const int* in_sizes, int n_in,
                              void* d_out, int out_size, void* d_ws, size_t ws_size,
                              hipStream_t stream) {
  (void)in_sizes; (void)n_in; (void)out_size; (void)ws_size;
  const float* x    = (const float*)d_in[0];   // [128,128]
  const float* W    = (const float*)d_in[1];   // [255,128]
  const float* bias = (const float*)d_in[2];   // [255]
  const float* L    = (const float*)d_in[3];   // [255,255]
  const float* G    = (const float*)d_in[4];   // [254,255]
  const float* z0   = (const float*)d_in[5];   // [255]
  const float* s0   = (const float*)d_in[6];   // [254]
  float* out        = (float*)d_out;           // [128,128]

  float* ws   = (float*)d_ws;
  float* Qg   = ws;                 // 256*256
  float* Gp   = Qg + NP * NP;       // 256*256
  float* GT   = Gp + NP * NP;       // 256*256
  float* pg   = GT + NP * NP;       // 128*256
  float* zg   = pg + NB * NP;       // 128*256
  float* sg   = zg + NB * NP;       // 128*256
  float* lamg = sg + NB * NP;       // 128*256
  float* hg   = lamg + NB * NP;     // 256

  const size_t SMEM = (size_t)(KST * NP + 11 * NP) * sizeof(float); // ~274 KB

  k_init  <<<NB, NP, 0, stream>>>(zg, sg, lamg);
  k_qbuild<<<NP, NP, 0, stream>>>(L, Qg);
  k_gbuild<<<NP, NP, 0, stream>>>(G, Gp, GT);
  k_pbuild<<<NB, NP, 0, stream>>>(x, W, bias, pg);
  k_hbuild<<<1,  NP, 0, stream>>>(G, z0, s0, hg);

  for (int it = 0; it < NITER; ++it)
    k_iter<<<NB, NP, SMEM, stream>>>(zg, sg, lamg, pg, Qg, Gp, GT, hg);

  k_out<<<NB, NF, 0, stream>>>(zg, out);
}